// DeltaNetLayer_79302276153498
// MI455X (gfx1250) — compile-verified
//
#include <hip/hip_runtime.h>
#include <hip/hip_bf16.h>

typedef float v2f __attribute__((ext_vector_type(2)));
typedef float v8f __attribute__((ext_vector_type(8)));

#define Bsz 16
#define Tlen 1024
#define Dd 256
#define Hh 64
#define CH 16      // chunk length for the WY-form scan
#define RPW 32     // state rows per scan workgroup (8 row-blocks per batch)

static __device__ __forceinline__ v8f wmma4(v2f a, v2f b, v8f c) {
    // D = A(16x4,f32) * B(4x16,f32) + C(16x16,f32)
    return __builtin_amdgcn_wmma_f32_16x16x4_f32(
        /*neg_a=*/false, a, /*neg_b=*/false, b,
        /*c_mod=*/(short)0, c, /*reuse_a=*/false, /*reuse_b=*/false);
}

// -------------------------------------------------------------------------
// Kernel 1: Y[M,N] = X[M,K] @ W[K,N] + bias   (fp32 WMMA 16x16x4)
// one wave per 16x16 output tile, 8 waves per block
// -------------------------------------------------------------------------
__global__ __launch_bounds__(256) void proj_gemm_kernel(
    const float* __restrict__ X, const float* __restrict__ W,
    const float* __restrict__ bias, float* __restrict__ Y,
    int M, int N, int K)
{
    const int wave = threadIdx.x >> 5;
    const int lane = threadIdx.x & 31;
    const int m = lane & 15;
    const int h = lane >> 4;           // lane half selects K pair
    const int ntilesN = N >> 4;
    const int tile = blockIdx.x * 8 + wave;
    const int mt = tile / ntilesN;
    const int nt = tile - mt * ntilesN;
    if (mt >= (M >> 4)) return;        // grid sized exactly; never taken

    const float* arow = X + (size_t)(mt * 16 + m) * K + 2 * h;        // A[m][kk+2h]
    const float* bcol = W + (size_t)(2 * h) * N + nt * 16 + m;        // B[kk+2h][n]
    v8f acc = {0.f, 0.f, 0.f, 0.f, 0.f, 0.f, 0.f, 0.f};
    for (int kk = 0; kk < K; kk += 4) {
        v2f a; a.x = arow[0]; a.y = arow[1];
        v2f b; b.x = bcol[0]; b.y = bcol[N];
        acc = wmma4(a, b, acc);
        arow += 4;
        bcol += 4 * (size_t)N;
    }
    const float bs = bias[nt * 16 + m];
    float* yout = Y + (size_t)(mt * 16 + 8 * h) * N + nt * 16 + m;    // rows p (h=0) / p+8 (h=1)
#pragma unroll
    for (int p = 0; p < 8; ++p)
        yout[(size_t)p * N] = acc[p] + bs;
}

// -------------------------------------------------------------------------
// Kernel 2: beta gate  sigmoid(relu(x@Wb1+bb1)@Wb2+bb2), one thread/token
// -------------------------------------------------------------------------
__global__ __launch_bounds__(256) void beta_gate_kernel(
    const float* __restrict__ x, const float* __restrict__ Wb1,
    const float* __restrict__ bb1, const float* __restrict__ Wb2,
    const float* __restrict__ bb2, float* __restrict__ beta, int ntok)
{
    const int tok = blockIdx.x * blockDim.x + threadIdx.x;
    if (tok >= ntok) return;
    const float* xr = x + (size_t)tok * Dd;
    float acc[Hh];
#pragma unroll
    for (int hh = 0; hh < Hh; ++hh) acc[hh] = bb1[hh];
    for (int d = 0; d < Dd; ++d) {
        const float xv = xr[d];
        const float* wr = Wb1 + (size_t)d * Hh;
#pragma unroll
        for (int hh = 0; hh < Hh; ++hh) acc[hh] = fmaf(xv, wr[hh], acc[hh]);
    }
    float z = bb2[0];
#pragma unroll
    for (int hh = 0; hh < Hh; ++hh) z = fmaf(fmaxf(acc[hh], 0.f), Wb2[hh], z);
    beta[tok] = 1.f / (1.f + __expf(-z));
}

// -------------------------------------------------------------------------
// Kernel 2b: per-chunk Gram matrices, shared by all row-block scan WGs:
//   gram[b][c][0][t][s] = k_t . k_s   (G)
//   gram[b][c][1][t][s] = q_t . k_s   (Aqk)
// one wave per 16x16 Gram (64 WMMA each); jobs = B * nch * 2
// -------------------------------------------------------------------------
__global__ __launch_bounds__(256) void gram_kernel(
    const float* __restrict__ qb, const float* __restrict__ kb,
    float* __restrict__ gram, int T, int nch)
{
    const int wave = threadIdx.x >> 5;
    const int lane = threadIdx.x & 31;
    const int m = lane & 15;
    const int h = lane >> 4;
    const int job = blockIdx.x * 8 + wave;           // 0 .. B*nch*2-1
    const int b = job / (nch * 2);
    const int rem = job - b * nch * 2;
    const int c = rem >> 1;
    const int which = rem & 1;                        // 0 = G, 1 = Aqk
    const size_t base = ((size_t)b * T + (size_t)c * CH) * Dd;
    const float* ar = (which ? qb : kb) + base + (size_t)m * Dd + 2 * h;  // A[m][kk]
    const float* br = kb + base + (size_t)m * Dd + 2 * h;                 // B[kk][n]=Kc[n][kk]
    v8f acc = {0.f, 0.f, 0.f, 0.f, 0.f, 0.f, 0.f, 0.f};
    for (int kk = 0; kk < Dd; kk += 4) {
        v2f a;  a.x = ar[kk];  a.y = ar[kk + 1];
        v2f bv; bv.x = br[kk]; bv.y = br[kk + 1];
        acc = wmma4(a, bv, acc);
    }
    float* gout = gram + ((size_t)(b * nch + c) * 2 + which) * 256;
#pragma unroll
    for (int p = 0; p < 8; ++p) gout[(p + 8 * h) * 16 + m] = acc[p];
}

// -------------------------------------------------------------------------
// Kernel 3: row-split chunked DeltaNet scan.
// Grid = B * (D/RPW) = 16*8 = 128 workgroups, 128 threads (4 wave32) each.
// Each WG owns a 32x256 slab of the state (exploits row-separability of the
// delta rule), kept in 32KB LDS with diagonal skew. Per chunk of 16 steps:
//   Phase A: waves 0-1: P  = Kc @ Sblk^T ; waves 2-3: PQ0 = Qc @ Sblk^T (WMMA)
//   Phase B: u_t = b_t (v_t - P_t - sum_{s<t} G[t,s] u_s)  (serial, 32 lanes)
//   Phase C: Sblk += U^T @ Kc                              (WMMA, in LDS)
//   Phase E: o_t = PQ0_t + sum_{s<=t} Aqk[t,s] u_s         (VALU)
// G and Aqk come precomputed from gram_kernel.
// -------------------------------------------------------------------------
#define SWZ(il, j) ((il) * 256 + (((il) + (j)) & 255))

__global__ __launch_bounds__(128) void deltanet_scan_kernel(
    const float* __restrict__ S0, const float* __restrict__ qb,
    const float* __restrict__ kb, const float* __restrict__ vb,
    const float* __restrict__ betab, const float* __restrict__ gram,
    float* __restrict__ outraw, float* __restrict__ Sfinal, int T)
{
    __shared__ float Sb[RPW * 256];   // 32KB state slab, skewed
    __shared__ float Pb[CH * RPW];    // P   [t][i]
    __shared__ float PQ[CH * RPW];    // PQ0 [t][i]
    __shared__ float Ub[CH * RPW];    // U   [t][i]
    __shared__ float GA[512];         // G (256) then Aqk (256)
    float* Gs = GA;
    float* AQ = GA + 256;

    const int b   = blockIdx.x >> 3;
    const int blk = blockIdx.x & 7;
    const int tid = threadIdx.x;
    const int wave = tid >> 5;
    const int lane = tid & 31;
    const int m = lane & 15;
    const int h = lane >> 4;
    const size_t base = (size_t)b * T * Dd;
    const int nch = T / CH;

    // load this WG's S0 rows [blk*RPW, +RPW) into skewed LDS
    for (int idx = tid; idx < RPW * 256; idx += 128) {
        const int il = idx >> 8, j = idx & 255;
        Sb[SWZ(il, j)] = S0[(size_t)b * (Dd * Dd) + (size_t)(blk * RPW + il) * 256 + j];
    }
    __syncthreads();

    for (int c = 0; c < nch; ++c) {
        const int t0 = c * CH;
        const float* Kc = kb + base + (size_t)t0 * Dd;
        const float* Qc = qb + base + (size_t)t0 * Dd;
        const float* Vc = vb + base + (size_t)t0 * Dd;

        // stage precomputed G / Aqk for this chunk
        const float* gsrc = gram + (size_t)((b * nch + c) * 2) * 256;
        for (int idx = tid; idx < 512; idx += 128) GA[idx] = gsrc[idx];

        if (c + 1 < nch)   // warm L1 for next chunk's K
            __builtin_prefetch(kb + base + (size_t)(t0 + CH) * Dd + tid * 32, 0, 0);

        // ---- Phase A: all 4 waves do one 16x16 WMMA tile vs Sblk^T
        {
            const int isQ = wave >> 1;           // waves 2,3 -> PQ0
            const int nt  = wave & 1;
            const int n   = nt * 16 + m;         // local state row 0..31
            const float* ar = (isQ ? Qc : Kc) + (size_t)m * Dd + 2 * h;
            v8f acc = {0.f, 0.f, 0.f, 0.f, 0.f, 0.f, 0.f, 0.f};
            for (int kk = 0; kk < Dd; kk += 4) {
                const int k0 = kk + 2 * h;
                v2f a;  a.x = ar[kk]; a.y = ar[kk + 1];
                v2f bv; bv.x = Sb[SWZ(n, k0)]; bv.y = Sb[SWZ(n, k0 + 1)];
                acc = wmma4(a, bv, acc);
            }
            float* dst = isQ ? PQ : Pb;
#pragma unroll
            for (int p = 0; p < 8; ++p) dst[(p + 8 * h) * RPW + n] = acc[p];
        }
        __syncthreads();

        // ---- Phase B: triangular solve for U (wave 0; lane = local row i)
        if (tid < RPW) {
            const int i = tid;
            const int gi = blk * RPW + i;
            float ul[CH];
#pragma unroll
            for (int t = 0; t < CH; ++t) {
                float val = Vc[t * Dd + gi] - Pb[t * RPW + i];
                for (int s = 0; s < t; ++s) val = fmaf(-Gs[t * 16 + s], ul[s], val);
                val *= betab[(size_t)b * T + t0 + t];
                ul[t] = val;
                Ub[t * RPW + i] = val;
            }
        }
        __syncthreads();

        // ---- Phase C: Sblk += U^T @ Kc   (2 mtiles x 16 ntiles, 8 tiles/wave)
        for (int j2 = 0; j2 < 8; ++j2) {
            const int tile = wave + 4 * j2;
            const int mt  = tile >> 4;
            const int nt2 = tile & 15;
            const int ia  = mt * 16 + m;     // local row for A fragment
            const int n   = nt2 * 16 + m;    // state column j
            v8f acc;
#pragma unroll
            for (int p = 0; p < 8; ++p) {
                const int il = mt * 16 + p + 8 * h;
                acc[p] = Sb[SWZ(il, n)];
            }
#pragma unroll
            for (int kk = 0; kk < CH; kk += 4) {
                v2f a;  a.x = Ub[(kk + 2 * h) * RPW + ia];
                        a.y = Ub[(kk + 2 * h + 1) * RPW + ia];
                v2f bv; bv.x = Kc[(size_t)(kk + 2 * h) * Dd + n];
                        bv.y = Kc[(size_t)(kk + 2 * h + 1) * Dd + n];
                acc = wmma4(a, bv, acc);
            }
#pragma unroll
            for (int p = 0; p < 8; ++p) {
                const int il = mt * 16 + p + 8 * h;
                Sb[SWZ(il, n)] = acc[p];
            }
        }

        // ---- Phase E (no barrier needed vs C: disjoint data):
        // o_t[gi] = PQ0[t][i] + sum_{s<=t} Aqk[t][s] * U[s][i]
        {
            const int i  = tid & 31;
            const int tq = tid >> 5;
            const int gi = blk * RPW + i;
#pragma unroll
            for (int tt = 0; tt < 4; ++tt) {
                const int t = tq * 4 + tt;
                float val = PQ[t * RPW + i];
                for (int s = 0; s <= t; ++s)
                    val = fmaf(AQ[t * 16 + s], Ub[s * RPW + i], val);
                outraw[base + (size_t)(t0 + t) * Dd + gi] = val;
            }
        }
        __syncthreads();   // protects Sb (A vs C) and GA/Ub across chunks
    }

    // final state slab
    for (int idx = tid; idx < RPW * 256; idx += 128) {
        const int il = idx >> 8, j = idx & 255;
        Sfinal[(size_t)b * (Dd * Dd) + (size_t)(blk * RPW + il) * 256 + j] = Sb[SWZ(il, j)];
    }
}

// -------------------------------------------------------------------------
// Kernel 4: in-place LayerNorm over last dim (one wave32 per row)
// -------------------------------------------------------------------------
__global__ __launch_bounds__(256) void layernorm_kernel(
    float* __restrict__ out, const float* __restrict__ gamma,
    const float* __restrict__ beta, int rows)
{
    const int wave = threadIdx.x >> 5;
    const int lane = threadIdx.x & 31;
    const int r = blockIdx.x * 8 + wave;
    if (r >= rows) return;
    float* row = out + (size_t)r * Dd;
    float v[8], s = 0.f, s2 = 0.f;
#pragma unroll
    for (int u = 0; u < 8; ++u) {
        v[u] = row[lane + 32 * u];
        s += v[u];
        s2 = fmaf(v[u], v[u], s2);
    }
    for (int off = 16; off; off >>= 1) {
        s  += __shfl_xor(s, off, 32);
        s2 += __shfl_xor(s2, off, 32);
    }
    const float mu = s * (1.f / Dd);
    const float var = s2 * (1.f / Dd) - mu * mu;
    const float rs = rsqrtf(var + 1e-5f);
#pragma unroll
    for (int u = 0; u < 8; ++u) {
        const int c = lane + 32 * u;
        row[c] = fmaf((v[u] - mu) * rs, gamma[c], beta[c]);
    }
}

// -------------------------------------------------------------------------
extern "C" void kernel_launch(void* const* d_in, const int* in_sizes, int n_in,
                              void* d_out, int out_size, void* d_ws, size_t ws_size,
                              hipStream_t stream) {
    const float* x     = (const float*)d_in[0];
    const float* S0    = (const float*)d_in[1];
    const float* Wq    = (const float*)d_in[2];
    const float* bq    = (const float*)d_in[3];
    const float* Wk    = (const float*)d_in[4];
    const float* bk    = (const float*)d_in[5];
    const float* Wv    = (const float*)d_in[6];
    const float* bv    = (const float*)d_in[7];
    const float* Wb1   = (const float*)d_in[8];
    const float* bb1   = (const float*)d_in[9];
    const float* Wb2   = (const float*)d_in[10];
    const float* bb2   = (const float*)d_in[11];
    const float* gamma = (const float*)d_in[12];
    const float* betaL = (const float*)d_in[13];

    const size_t ntok = (size_t)Bsz * Tlen;        // 16384
    const size_t nQKV = ntok * Dd;                 // 4,194,304 floats each
    const int    nch  = Tlen / CH;                 // 64 chunks

    float* ws   = (float*)d_ws;
    float* qbuf = ws;
    float* kbuf = qbuf + nQKV;
    float* vbuf = kbuf + nQKV;
    float* bbuf = vbuf + nQKV;                     // [ntok]
    float* gbuf = bbuf + ntok;                     // [B][nch][2][16][16]

    float* out  = (float*)d_out;                   // [B,T,D]
    float* Sfin = out + nQKV;                      // [B,D,D]

    // projections: 16384x16 tiles, 8 waves/block -> 2048 blocks
    proj_gemm_kernel<<<2048, 256, 0, stream>>>(x, Wq, bq, qbuf, (int)ntok, Dd, Dd);
    proj_gemm_kernel<<<2048, 256, 0, stream>>>(x, Wk, bk, kbuf, (int)ntok, Dd, Dd);
    proj_gemm_kernel<<<2048, 256, 0, stream>>>(x, Wv, bv, vbuf, (int)ntok, Dd, Dd);
    beta_gate_kernel<<<64, 256, 0, stream>>>(x, Wb1, bb1, Wb2, bb2, bbuf, (int)ntok);

    // per-chunk Grams: B*nch*2 = 2048 wave-jobs -> 256 blocks of 8 waves
    gram_kernel<<<256, 256, 0, stream>>>(qbuf, kbuf, gbuf, Tlen, nch);

    // scan: 16 batches x 8 row-blocks = 128 WGs, 4 wave32 each, 40KB LDS
    deltanet_scan_kernel<<<Bsz * 8, 128, 0, stream>>>(S0, qbuf, kbuf, vbuf, bbuf,
                                                      gbuf, out, Sfin, Tlen);

    layernorm_kernel<<<2048, 256, 0, stream>>>(out, gamma, betaL, (int)ntok);
}